// UncertaintyAwareIoULoss_55980603736214
// MI455X (gfx1250) — compile-verified
//
#include <hip/hip_runtime.h>
#include <hip/hip_bf16.h>
#include <math.h>

typedef __attribute__((ext_vector_type(2))) float v2f;
typedef __attribute__((ext_vector_type(8))) float v8f;

#define RES 128
#define NPTS 192
#define TOPK 16
#define NCOMBO 4
#define PB_STRIDE 64
#define PARAM_BASE 8
#define PART_OFF 384
#define NBLOCKS 1024
#define WPB 8                                      // waves per block (wave32)
#define NTILES ((RES * RES * RES) / 16)            // 131072 tiles of 16 samples
#define TILES_PER_WAVE (NTILES / (NBLOCKS * WPB))  // 16

// exp(-d2/(2*sigma^2)) computed as exp2(C2*d2), C2 = -log2(e)/(2*sigma^2)
#define SIGMA_F (6.0f / (float)RES)
#define C2_F (-1.4426950408889634f / (2.0f * SIGMA_F * SIGMA_F))

__device__ __forceinline__ float fast_exp2(float x) {
#if __has_builtin(__builtin_amdgcn_exp2f)
  return __builtin_amdgcn_exp2f(x);
#else
  return exp2f(x);
#endif
}

// ---------------------------------------------------------------------------
// Kernel 1: top-k select + renormalize + gather + grid bounds + A-matrix bake.
// One wave per (set, batch) combo; combo s: set = s>>1, b = s&1.
// Workspace floats: [0..2]=lo, [3..5]=step, then per combo s at 8 + s*64:
//   a0[16] = -2c*px, a1[16] = -2c*py, a2[16] = -2c*pz,
//   a3[16] = c*|p|^2 + log2(w_norm)
// ---------------------------------------------------------------------------
__global__ __launch_bounds__(128) void prep_kernel(
    const float* __restrict__ pts1, const float* __restrict__ w1,
    const float* __restrict__ pts2, const float* __restrict__ w2,
    float* __restrict__ ws) {
  const int tid  = threadIdx.x;
  const int s    = tid >> 5;
  const int lane = tid & 31;
  const int b    = s & 1;
  const float* wrow = ((s < 2) ? w1 : w2) + b * NPTS;
  const float* prow = ((s < 2) ? pts1 : pts2) + b * NPTS * 3;

  __shared__ float sv[NCOMBO][TOPK];
  __shared__ int   si[NCOMBO][TOPK];
  __shared__ float smin[NCOMBO][3];
  __shared__ float smax[NCOMBO][3];

  float vals[6];
#pragma unroll
  for (int t = 0; t < 6; ++t) vals[t] = wrow[lane + 32 * t];

  float sumv = 0.0f;
  for (int it = 0; it < TOPK; ++it) {
    float bv = -INFINITY;
    int bi = 0x7fffffff;
#pragma unroll
    for (int t = 0; t < 6; ++t) {
      float v = vals[t];
      int ii = lane + 32 * t;
      if (v > bv) { bv = v; bi = ii; }
    }
    // wave-wide argmax, tie -> lowest index (matches lax.top_k ordering)
#pragma unroll
    for (int off = 16; off >= 1; off >>= 1) {
      float ov = __shfl_xor(bv, off);
      int   oi = __shfl_xor(bi, off);
      if (ov > bv || (ov == bv && oi < bi)) { bv = ov; bi = oi; }
    }
    sumv += bv;
    if (lane == 0) { sv[s][it] = bv; si[s][it] = bi; }
    if ((bi & 31) == lane) vals[bi >> 5] = -INFINITY;
  }
  __syncthreads();

  float px = 0.f, py = 0.f, pz = 0.f;
  if (lane < TOPK) {
    float v = sv[s][lane];
    int   i = si[s][lane];
    float wn = v / (sumv + 1e-5f);
    px = prow[i * 3 + 0];
    py = prow[i * 3 + 1];
    pz = prow[i * 3 + 2];
    const int base = PARAM_BASE + s * PB_STRIDE;
    const float pn2 = px * px + py * py + pz * pz;
    ws[base + lane]      = (-2.0f * C2_F) * px;
    ws[base + 16 + lane] = (-2.0f * C2_F) * py;
    ws[base + 32 + lane] = (-2.0f * C2_F) * pz;
    ws[base + 48 + lane] = C2_F * pn2 + log2f(wn);
  }
  float mnx = px, mny = py, mnz = pz, mxx = px, mxy = py, mxz = pz;
  if (lane >= TOPK) {
    mnx = mny = mnz = INFINITY;
    mxx = mxy = mxz = -INFINITY;
  }
#pragma unroll
  for (int off = 8; off >= 1; off >>= 1) {  // masks <16: stays within lanes 0..15
    mnx = fminf(mnx, __shfl_xor(mnx, off));
    mny = fminf(mny, __shfl_xor(mny, off));
    mnz = fminf(mnz, __shfl_xor(mnz, off));
    mxx = fmaxf(mxx, __shfl_xor(mxx, off));
    mxy = fmaxf(mxy, __shfl_xor(mxy, off));
    mxz = fmaxf(mxz, __shfl_xor(mxz, off));
  }
  if (lane == 0) {
    smin[s][0] = mnx; smin[s][1] = mny; smin[s][2] = mnz;
    smax[s][0] = mxx; smax[s][1] = mxy; smax[s][2] = mxz;
  }
  __syncthreads();
  if (tid == 0) {
    const float pad = 3.0f * SIGMA_F;
#pragma unroll
    for (int a = 0; a < 3; ++a) {
      float lo = INFINITY, hi = -INFINITY;
      for (int c = 0; c < NCOMBO; ++c) {
        lo = fminf(lo, smin[c][a]);
        hi = fmaxf(hi, smax[c][a]);
      }
      lo -= pad;
      hi += pad;
      ws[a]     = lo;
      ws[3 + a] = (hi - lo) / (float)(RES - 1);
    }
  }
}

// ---------------------------------------------------------------------------
// Kernel 2: one V_WMMA_F32_16X16X4_F32 per mixture per 16-sample tile.
//   A (rows = points, loop-invariant):  [-2c*px, -2c*py, -2c*pz, c|p|^2+log2 w]
//   B (cols = samples, built per tile): [x, y, z, 1]
//   D[p][n] + c|s_n|^2 = c*d2 + log2(w)  ->  exp2 = w * gaussian
// C/D layout: lane = sample n, VGPR r = point p (low half p=r, high half p=r+8)
// => per-lane sum over 8 regs. Low half holds sum over points 0..7, high half
// points 8..15. Sums Σh1, Σh2 are accumulated on all lanes (wave reduction at
// the end combines halves). The product term uses only 2 shfl_xor(16)/tile:
//   Σ_lanes p0*(p2+o2) + p1*(p3+o3) == Σ_n h0*h2 + h1*h3   (exact)
// No EXEC divergence in the hot loop; LDS traffic = 2 permutes per tile.
// ---------------------------------------------------------------------------
__global__ __launch_bounds__(256) void heat_kernel(const float* __restrict__ ws,
                                                   float* __restrict__ partials) {
  const int lane = threadIdx.x & 31;
  const int wave = threadIdx.x >> 5;
  const int n    = lane & 15;
  const bool hih = lane >= 16;

  // A matrices: lane = row p; low half rows K0/K1, high half rows K2/K3.
  v2f A[NCOMBO];
#pragma unroll
  for (int s = 0; s < NCOMBO; ++s) {
    const float* pb = ws + PARAM_BASE + s * PB_STRIDE;
    A[s] = (v2f){hih ? pb[32 + n] : pb[n],        // K0=-2c*px | K2=-2c*pz
                 hih ? pb[48 + n] : pb[16 + n]};  // K1=-2c*py | K3=c|p|^2+log2w
  }

  const float lo0 = ws[0], lo1 = ws[1], lo2 = ws[2];
  const float st0 = ws[3], st1 = ws[4], st2 = ws[5];

  float accI = 0.f, acc1 = 0.f, acc2 = 0.f;
  const int gw = blockIdx.x * WPB + wave;
  const int t0 = gw * TILES_PER_WAVE;

  for (int t = t0; t < t0 + TILES_PER_WAVE; ++t) {
    const int id = t * 16 + n;
    const int i = id >> 14, j = (id >> 7) & 127, k = id & 127;
    const float x = lo0 + (float)i * st0;
    const float y = lo1 + (float)j * st1;
    const float z = lo2 + (float)k * st2;
    const float csn2 = C2_F * (x * x + y * y + z * z);  // per-lane column term
    // B: lane = col n (this lane's sample); low rows K0/K1, high rows K2/K3.
    v2f B = (v2f){hih ? z : x, hih ? 1.0f : y};
    v8f zacc = {};

    float p[NCOMBO];
#pragma unroll
    for (int s = 0; s < NCOMBO; ++s) {
      v8f c = __builtin_amdgcn_wmma_f32_16x16x4_f32(
          false, A[s], false, B, (short)0, zacc, false, false);
      float partial = 0.f;
#pragma unroll
      for (int r = 0; r < 8; ++r) partial += fast_exp2(c[r] + csn2);
      p[s] = partial;  // low half: points 0..7; high half: points 8..15
    }
    const float o2 = __shfl_xor(p[2], 16);
    const float o3 = __shfl_xor(p[3], 16);
    accI += p[0] * (p[2] + o2) + p[1] * (p[3] + o3);
    acc1 += p[0] + p[1];
    acc2 += p[2] + p[3];
  }

#pragma unroll
  for (int off = 16; off >= 1; off >>= 1) {
    accI += __shfl_xor(accI, off);
    acc1 += __shfl_xor(acc1, off);
    acc2 += __shfl_xor(acc2, off);
  }
  __shared__ float red[WPB][3];
  if (lane == 0) {
    red[wave][0] = accI;
    red[wave][1] = acc1;
    red[wave][2] = acc2;
  }
  __syncthreads();
  if (threadIdx.x == 0) {
    float I = 0.f, s1 = 0.f, s2 = 0.f;
    for (int w = 0; w < WPB; ++w) {
      I += red[w][0];
      s1 += red[w][1];
      s2 += red[w][2];
    }
    partials[blockIdx.x * 3 + 0] = I;
    partials[blockIdx.x * 3 + 1] = s1;
    partials[blockIdx.x * 3 + 2] = s2;
  }
}

// ---------------------------------------------------------------------------
// Kernel 3: deterministic final reduction + IoU.
// ---------------------------------------------------------------------------
__global__ __launch_bounds__(256) void finish_kernel(const float* __restrict__ partials,
                                                     float* __restrict__ out) {
  __shared__ float rI[256], r1[256], r2[256];
  float I = 0.f, s1 = 0.f, s2 = 0.f;
  for (int t = threadIdx.x; t < NBLOCKS; t += 256) {
    I  += partials[t * 3 + 0];
    s1 += partials[t * 3 + 1];
    s2 += partials[t * 3 + 2];
  }
  rI[threadIdx.x] = I;
  r1[threadIdx.x] = s1;
  r2[threadIdx.x] = s2;
  __syncthreads();
  for (int off = 128; off >= 1; off >>= 1) {
    if (threadIdx.x < off) {
      rI[threadIdx.x] += rI[threadIdx.x + off];
      r1[threadIdx.x] += r1[threadIdx.x + off];
      r2[threadIdx.x] += r2[threadIdx.x + off];
    }
    __syncthreads();
  }
  if (threadIdx.x == 0) {
    float U = r1[0] + r2[0] - rI[0] + 1e-5f;
    out[0] = 1.0f - rI[0] / U;
  }
}

extern "C" void kernel_launch(void* const* d_in, const int* in_sizes, int n_in,
                              void* d_out, int out_size, void* d_ws, size_t ws_size,
                              hipStream_t stream) {
  (void)in_sizes; (void)n_in; (void)out_size; (void)ws_size;
  const float* pts1 = (const float*)d_in[0];
  const float* w1   = (const float*)d_in[1];
  const float* pts2 = (const float*)d_in[2];
  const float* w2   = (const float*)d_in[3];
  float* ws = (float*)d_ws;
  float* partials = ws + PART_OFF;

  prep_kernel<<<1, 128, 0, stream>>>(pts1, w1, pts2, w2, ws);
  heat_kernel<<<NBLOCKS, 256, 0, stream>>>(ws, partials);
  finish_kernel<<<1, 256, 0, stream>>>(partials, (float*)d_out);
}